// DiscreteNet_45028437131792
// MI455X (gfx1250) — compile-verified
//
#include <hip/hip_runtime.h>
#include <hip/hip_bf16.h>
#include <math.h>

#define Bq 32
#define Tt 128
#define Dd 3072
#define NVv 4
#define CSs 6
#define Sz 1296
#define Av 4
#define Lu 5
#define BT (Bq*Tt)
#define KT4 (Sz/4)      /* 324 k-tiles of 4 for T matrices */
#define CT  (Sz/16)     /* 81 col tiles */

typedef float v2f __attribute__((ext_vector_type(2)));
typedef float v8f __attribute__((ext_vector_type(8)));

// ---- workspace layout (float offsets) ----
#define OFF_LOGITS ((size_t)0)                                  // BT*Sz (becomes ol in place)
#define OFF_LOGLAT (OFF_LOGITS + (size_t)BT*Sz)                 // BT*24
#define OFF_TPACK  (OFF_LOGLAT + (size_t)BT*NVv*CSs)            // Av*Sz*Sz  (B-fragment packed softmax(T))
#define OFF_WDPACK (OFF_TPACK  + (size_t)Av*Sz*Sz)              // Dd*Sz     (B-fragment packed W_dec)
#define OFF_POST   (OFF_WDPACK + (size_t)Dd*Sz)                 // Tt*Bq*Sz
#define OFF_PRIORB (OFF_POST   + (size_t)Tt*Bq*Sz)              // Sz
#define OFF_ACC    (OFF_PRIORB + (size_t)Sz)                    // 4 floats: recon, latent, prior, dyn

__device__ inline void wgp_prefetch(const void* p) {
    // default scope field = 0 -> SCOPE_WGP: pull into all cache levels on miss
    asm volatile("global_prefetch_b8 %0, off" :: "v"(p));
}

// ---- block reductions (blockDim must be power of two) ----
__device__ inline float blkSum(float v, float* red) {
    int tid = threadIdx.x;
    red[tid] = v; __syncthreads();
    for (int off = blockDim.x >> 1; off > 0; off >>= 1) {
        if (tid < off) red[tid] += red[tid + off];
        __syncthreads();
    }
    float r = red[0]; __syncthreads();
    return r;
}
__device__ inline float blkMax(float v, float* red) {
    int tid = threadIdx.x;
    red[tid] = v; __syncthreads();
    for (int off = blockDim.x >> 1; off > 0; off >>= 1) {
        if (tid < off) red[tid] = fmaxf(red[tid], red[tid + off]);
        __syncthreads();
    }
    float r = red[0]; __syncthreads();
    return r;
}

// ---- K0: prior_b = softmax(prior_logits) ----
__global__ void k_prior_softmax(const float* __restrict__ pl, float* __restrict__ priorb) {
    __shared__ float red[512];
    int tid = threadIdx.x;
    float x[3]; float m = -1e30f;
    #pragma unroll
    for (int i = 0; i < 3; ++i) {
        int e = tid + i * 512;
        x[i] = (e < Sz) ? pl[e] : -1e30f;
        m = fmaxf(m, x[i]);
    }
    m = blkMax(m, red);
    float s = 0.f;
    #pragma unroll
    for (int i = 0; i < 3; ++i) { int e = tid + i * 512; if (e < Sz) s += expf(x[i] - m); }
    s = blkSum(s, red);
    float inv = 1.f / s;
    #pragma unroll
    for (int i = 0; i < 3; ++i) { int e = tid + i * 512; if (e < Sz) priorb[e] = expf(x[i] - m) * inv; }
}

// ---- K1: encoder GEMM (BT x 24), per-factor log-softmax, latent_loss ----
__global__ void k_enc(const float* __restrict__ obs, const float* __restrict__ Wenc,
                      float* __restrict__ loglat, float* __restrict__ accG) {
    __shared__ float red[128];
    __shared__ float encv[NVv * CSs];
    int row = blockIdx.x, tid = threadIdx.x;
    float part[24];
    #pragma unroll
    for (int j = 0; j < 24; ++j) part[j] = 0.f;
    const float* orow = obs + (size_t)row * Dd;
    for (int k = tid; k < Dd; k += 128) {
        float o = orow[k];
        const float* wr = Wenc + (size_t)k * 24;
        #pragma unroll
        for (int j = 0; j < 24; ++j) part[j] = fmaf(o, wr[j], part[j]);
    }
    for (int j = 0; j < 24; ++j) {
        red[tid] = part[j]; __syncthreads();
        for (int off = 64; off > 0; off >>= 1) {
            if (tid < off) red[tid] += red[tid + off];
            __syncthreads();
        }
        if (tid == 0) encv[j] = red[0];
        __syncthreads();
    }
    if (tid < NVv) {
        int v = tid;
        float m = -1e30f;
        for (int c = 0; c < CSs; ++c) m = fmaxf(m, encv[v * CSs + c]);
        float s = 0.f;
        for (int c = 0; c < CSs; ++c) s += expf(encv[v * CSs + c] - m);
        float ls = m + logf(s);
        float lat = 0.f;
        for (int c = 0; c < CSs; ++c) {
            float ll = encv[v * CSs + c] - ls;
            loglat[(size_t)row * 24 + v * CSs + c] = ll;
            lat += expf(ll) * ll;
        }
        red[tid] = lat;
    }
    __syncthreads();
    if (tid == 0) atomicAdd(&accG[1], red[0] + red[1] + red[2] + red[3]);
}

// ---- K2a: pack W_dec (D x S) into WMMA B-fragment order ----
// frag (kt, ct): 64 floats, lane = kh*16+mrow, pair j in {0,1}
// value = Wd[(4*kt + 2*kh + j)*Sz + ct*16 + mrow]
__global__ void k_wdpack(const float* __restrict__ Wd, float* __restrict__ pack) {
    size_t idx = (size_t)blockIdx.x * 256 + threadIdx.x;   // over Dd*Sz
    int k = (int)(idx / Sz), col = (int)(idx % Sz);
    int kt = k >> 2, kr = k & 3, kh = kr >> 1, j = kr & 1;
    int ct = col >> 4, mrow = col & 15;
    int lane = kh * 16 + mrow;
    pack[(((size_t)kt * CT + ct) * 64) + lane * 2 + j] = Wd[idx];
}

// ---- K2b: decoder GEMM logits[BT x Sz] = obs @ W_dec, f32 WMMA 16x16x4 ----
// wave tile = 16 rows x 48 cols; 256 row-tiles * 27 col-groups = 6912 waves = 864 blocks of 8 waves
__global__ void k_dec_gemm(const float* __restrict__ obs, const float* __restrict__ wdp,
                           float* __restrict__ logits) {
    int wid = blockIdx.x * 8 + (threadIdx.x >> 5);
    int lane = threadIdx.x & 31;
    int rowTile = wid / 27, colGroup = wid % 27;
    int mrow = lane & 15, kh = lane >> 4;
    const float* arow = obs + (size_t)(rowTile * 16 + mrow) * Dd;
    int ct0 = colGroup * 3;
    const float* bp0 = wdp + ((size_t)0 * CT + (ct0 + 0)) * 64 + lane * 2;
    const float* bp1 = wdp + ((size_t)0 * CT + (ct0 + 1)) * 64 + lane * 2;
    const float* bp2 = wdp + ((size_t)0 * CT + (ct0 + 2)) * 64 + lane * 2;
    v8f acc0 = {}, acc1 = {}, acc2 = {};
    for (int k0 = 0; k0 < Dd; k0 += 4) {
        v2f a = *(const v2f*)(arow + k0 + 2 * kh);
        size_t fo = (size_t)(k0 >> 2) * (CT * 64);
        wgp_prefetch(arow + k0 + 128);
        wgp_prefetch(bp0 + fo + 8 * (CT * 64));
        v2f b;
        b = *(const v2f*)(bp0 + fo);
        acc0 = __builtin_amdgcn_wmma_f32_16x16x4_f32(false, a, false, b, (short)0, acc0, false, false);
        b = *(const v2f*)(bp1 + fo);
        acc1 = __builtin_amdgcn_wmma_f32_16x16x4_f32(false, a, false, b, (short)0, acc1, false, false);
        b = *(const v2f*)(bp2 + fo);
        acc2 = __builtin_amdgcn_wmma_f32_16x16x4_f32(false, a, false, b, (short)0, acc2, false, false);
    }
    #pragma unroll
    for (int j = 0; j < 8; ++j) {
        int r = rowTile * 16 + j + 8 * kh;
        size_t base = (size_t)r * Sz + (size_t)colGroup * 48 + mrow;
        logits[base]      = acc0[j];
        logits[base + 16] = acc1[j];
        logits[base + 32] = acc2[j];
    }
}

// ---- K3: per-row log-softmax (in place -> ol) + recon via logsumexp(ol + lat_sum) ----
__global__ void k_rowops(float* __restrict__ logits, const float* __restrict__ loglat,
                         float* __restrict__ accG) {
    __shared__ float red[256];
    __shared__ float ll[24];
    int row = blockIdx.x, tid = threadIdx.x;
    float* lrow = logits + (size_t)row * Sz;
    float x[6]; float m = -1e30f;
    #pragma unroll
    for (int i = 0; i < 6; ++i) {
        int e = tid + i * 256;
        x[i] = (e < Sz) ? lrow[e] : -1e30f;
        m = fmaxf(m, x[i]);
    }
    m = blkMax(m, red);
    float s = 0.f;
    #pragma unroll
    for (int i = 0; i < 6; ++i) { int e = tid + i * 256; if (e < Sz) s += expf(x[i] - m); }
    s = blkSum(s, red);
    float ls = m + logf(s);
    #pragma unroll
    for (int i = 0; i < 6; ++i) {
        int e = tid + i * 256;
        if (e < Sz) { x[i] -= ls; lrow[e] = x[i]; }
    }
    if (tid < 24) ll[tid] = loglat[(size_t)row * 24 + tid];
    __syncthreads();
    float z[6]; float m2 = -1e30f;
    #pragma unroll
    for (int i = 0; i < 6; ++i) {
        int e = tid + i * 256;
        if (e < Sz) {
            int v0 = e / 216, v1 = (e / 36) % 6, v2 = (e / 6) % 6, v3 = e % 6;
            z[i] = x[i] + ll[v0] + ll[6 + v1] + ll[12 + v2] + ll[18 + v3];
        } else z[i] = -1e30f;
        m2 = fmaxf(m2, z[i]);
    }
    m2 = blkMax(m2, red);
    float s2 = 0.f;
    #pragma unroll
    for (int i = 0; i < 6; ++i) { int e = tid + i * 256; if (e < Sz) s2 += expf(z[i] - m2); }
    s2 = blkSum(s2, red);
    if (tid == 0) atomicAdd(&accG[0], -(m2 + logf(s2)));
}

// ---- K4: softmax(T_logits) written directly in WMMA B-fragment packed order ----
// one block per (a, k) row; row value at col e goes to frag ((a,kt),ct), lane, pair j
__global__ void k_tmat(const float* __restrict__ Tl, float* __restrict__ tpack) {
    __shared__ float red[256];
    int row = blockIdx.x;                 // 0 .. Av*Sz-1
    int a = row / Sz, k = row % Sz;
    int kt = k >> 2, kr = k & 3, kh = kr >> 1, j = kr & 1;
    const float* r = Tl + (size_t)row * Sz;
    float* base = tpack + (((size_t)a * KT4 + kt) * CT) * 64 + j;
    int tid = threadIdx.x;
    float x[6]; float m = -1e30f;
    #pragma unroll
    for (int i = 0; i < 6; ++i) {
        int e = tid + i * 256;
        x[i] = (e < Sz) ? r[e] : -1e30f;
        m = fmaxf(m, x[i]);
    }
    m = blkMax(m, red);
    float s = 0.f;
    #pragma unroll
    for (int i = 0; i < 6; ++i) { int e = tid + i * 256; if (e < Sz) s += expf(x[i] - m); }
    s = blkSum(s, red);
    float inv = 1.f / s;
    #pragma unroll
    for (int i = 0; i < 6; ++i) {
        int e = tid + i * 256;
        if (e < Sz) {
            int ct = e >> 4, mrow = e & 15;
            int lane = kh * 16 + mrow;
            base[(size_t)ct * 64 + lane * 2] = expf(x[i] - m) * inv;
        }
    }
}

// ---- K5: sequential posterior scan per batch row; also prior_loss at t=0 ----
__global__ void k_scan(const float* __restrict__ ol, const float* __restrict__ priorb,
                       float* __restrict__ post, float* __restrict__ accG) {
    __shared__ float red[512];
    int b = blockIdx.x, tid = threadIdx.x;
    float pb[3], p[3];
    #pragma unroll
    for (int i = 0; i < 3; ++i) { int e = tid + i * 512; pb[i] = (e < Sz) ? priorb[e] : 0.f; p[i] = 0.f; }
    const float* o0 = ol + (size_t)(b * Tt) * Sz;
    float sloc = 0.f;
    #pragma unroll
    for (int i = 0; i < 3; ++i) {
        int e = tid + i * 512;
        if (e < Sz) { p[i] = pb[i] * expf(o0[e]); sloc += p[i]; }
    }
    float tot = blkSum(sloc, red);
    float inv = 1.f / tot;
    float prloc = 0.f;
    #pragma unroll
    for (int i = 0; i < 3; ++i) {
        int e = tid + i * 512;
        if (e < Sz) {
            p[i] *= inv;
            post[(size_t)b * Sz + e] = p[i];   // t = 0
            prloc += pb[i] * (logf(fmaxf(pb[i], 1e-30f)) - logf(fmaxf(p[i], 1e-30f)));
        }
    }
    prloc = blkSum(prloc, red);
    if (tid == 0) atomicAdd(&accG[2], prloc);
    for (int t = 1; t < Tt; ++t) {
        const float* ot = ol + (size_t)(b * Tt + t) * Sz;
        sloc = 0.f;
        #pragma unroll
        for (int i = 0; i < 3; ++i) {
            int e = tid + i * 512;
            if (e < Sz) { p[i] = p[i] * expf(ot[e]) + 1e-10f; sloc += p[i]; }
        }
        tot = blkSum(sloc, red);
        inv = 1.f / tot;
        #pragma unroll
        for (int i = 0; i < 3; ++i) {
            int e = tid + i * 512;
            if (e < Sz) { p[i] *= inv; post[(size_t)(t * Bq + b) * Sz + e] = p[i]; }
        }
    }
}

// ---- K6: rollout chains with masked 4-action WMMA; dyn_loss KL at endpoint ----
// grid = (Tt-1)*2 blocks, 512 threads; dyn shared = (16*Sz + 16 + 512) floats
extern __shared__ float smem[];
__global__ void k_rollout(const float* __restrict__ post, const float* __restrict__ tpack,
                          const int* __restrict__ act, float* __restrict__ accG) {
    float* V = smem;                                   // 16 x Sz
    int* actS = (int*)(smem + 16 * Sz);                // 16
    float* red = smem + 16 * Sz + 16;                  // 512
    int t = blockIdx.x / 2 + 1;
    int half = blockIdx.x & 1;
    int tid = threadIdx.x;
    int wave = tid >> 5, lane = tid & 31, mrow = lane & 15, kh = lane >> 4;
    int s0 = t - Lu; if (s0 < 0) s0 = 0;
    int Lsteps = t - s0;

    // async copy posteriors[s0, 16 rows] -> LDS V (82944 B as b128 chunks, ASYNCcnt path)
    {
        const float* src = post + (size_t)(s0 * Bq + half * 16) * Sz;
        unsigned ldsBase = (unsigned)(uintptr_t)(void*)V;
        for (int c = tid; c < (16 * Sz) / 4; c += 512) {
            unsigned la = ldsBase + (unsigned)c * 16u;
            const float* g = src + (size_t)c * 4;
            asm volatile("global_load_async_to_lds_b128 %0, %1, off"
                         :: "v"(la), "v"(g) : "memory");
        }
        asm volatile("s_wait_asynccnt 0x0" ::: "memory");
    }
    __syncthreads();

    for (int j = 0; j < Lsteps; ++j) {
        int time = s0 + j;
        if (tid < 16) actS[tid] = act[(half * 16 + tid) * Tt + time];
        __syncthreads();
        v8f accT[6];
        int nt = 0;
        for (int tile = wave; tile < CT; tile += 16) {
            v8f c = {};
            const float* bbase = tpack + (size_t)tile * 64 + lane * 2;
            for (int k0 = 0; k0 < Sz; k0 += 4) {
                v2f a = *(const v2f*)(V + mrow * Sz + k0 + 2 * kh);
                size_t fo = (size_t)(k0 >> 2) * (CT * 64);
                #pragma unroll
                for (int aa = 0; aa < 4; ++aa) {
                    float msk = (actS[mrow] == aa) ? 1.f : 0.f;
                    v2f am; am.x = a.x * msk; am.y = a.y * msk;
                    v2f bv = *(const v2f*)(bbase + (size_t)aa * KT4 * (CT * 64) + fo);
                    c = __builtin_amdgcn_wmma_f32_16x16x4_f32(false, am, false, bv, (short)0, c, false, false);
                }
            }
            accT[nt++] = c;
        }
        __syncthreads();
        nt = 0;
        for (int tile = wave; tile < CT; tile += 16) {
            #pragma unroll
            for (int q = 0; q < 8; ++q)
                V[(q + 8 * kh) * Sz + tile * 16 + mrow] = accT[nt][q];
            nt++;
        }
        __syncthreads();
    }

    const float* postt = post + (size_t)(t * Bq + half * 16) * Sz;
    float acc = 0.f;
    for (int i = tid; i < 16 * Sz; i += 512) {
        float p = V[i];
        float q = postt[i];
        acc += p * (logf(fmaxf(p, 1e-30f)) - logf(fmaxf(q, 1e-30f)));
    }
    acc = blkSum(acc, red);
    if (tid == 0) atomicAdd(&accG[3], acc);
}

// ---- K7: finalize 5 scalars ----
__global__ void k_final(const float* __restrict__ accG, float* __restrict__ out) {
    if (threadIdx.x == 0 && blockIdx.x == 0) {
        out[0] = accG[0] / (float)BT;        // recon_loss
        out[1] = accG[1] / (float)BT;        // latent_loss
        out[2] = accG[2] / (float)Bq;        // prior_loss
        out[3] = 0.f;                        // value_prefix_loss
        out[4] = accG[3] / (float)(Bq * Tt); // dyn_loss
    }
}

extern "C" void kernel_launch(void* const* d_in, const int* in_sizes, int n_in,
                              void* d_out, int out_size, void* d_ws, size_t ws_size,
                              hipStream_t stream) {
    (void)in_sizes; (void)n_in; (void)out_size; (void)ws_size;
    const float* obs   = (const float*)d_in[0];
    const int*   act   = (const int*)d_in[1];
    // d_in[2] reward_sequence: unused (value_prefix_loss == 0)
    const float* prl   = (const float*)d_in[3];
    const float* Tl    = (const float*)d_in[4];
    const float* Wenc  = (const float*)d_in[5];
    const float* Wd    = (const float*)d_in[6];
    float* ws = (float*)d_ws;

    float* logits = ws + OFF_LOGITS;
    float* loglat = ws + OFF_LOGLAT;
    float* tpack  = ws + OFF_TPACK;
    float* wdpack = ws + OFF_WDPACK;
    float* postw  = ws + OFF_POST;
    float* priorb = ws + OFF_PRIORB;
    float* accG   = ws + OFF_ACC;

    hipMemsetAsync(accG, 0, 4 * sizeof(float), stream);

    k_prior_softmax<<<1, 512, 0, stream>>>(prl, priorb);
    k_enc<<<BT, 128, 0, stream>>>(obs, Wenc, loglat, accG);
    k_wdpack<<<(Dd * Sz) / 256, 256, 0, stream>>>(Wd, wdpack);
    k_dec_gemm<<<(256 * 27) / 8, 256, 0, stream>>>(obs, wdpack, logits);
    k_rowops<<<BT, 256, 0, stream>>>(logits, loglat, accG);
    k_tmat<<<Av * Sz, 256, 0, stream>>>(Tl, tpack);
    k_scan<<<Bq, 512, 0, stream>>>(logits, priorb, postw, accG);
    size_t shmem = (size_t)(16 * Sz + 16 + 512) * sizeof(float);
    k_rollout<<<(Tt - 1) * 2, 512, shmem, stream>>>(postw, tpack, act, accG);
    k_final<<<1, 32, 0, stream>>>(accG, (float*)d_out);
}